// BradleyTerryModel_74337293959225
// MI455X (gfx1250) — compile-verified
//
#include <hip/hip_runtime.h>
#include <hip/hip_bf16.h>

typedef float v2f __attribute__((ext_vector_type(2)));
typedef float v8f __attribute__((ext_vector_type(8)));

#define NDIM 8192
#define ROWS_PER_BLOCK 16
#define THREADS 256
#define GRID (NDIM / ROWS_PER_BLOCK)   // 512 blocks
#define COLS_PER_CHUNK (THREADS * 4)   // 1024 columns per chunk

#define LOG2E 1.4426950408889634f
#define LN2   0.6931471805599453f

// -(beta_i - logaddexp(beta_i, beta_j)) >= 0, computed stably in base-2 HW ops.
__device__ __forceinline__ float neg_logp(float bi, float bj) {
    float d = bj - bi;
    // 2^(-|d|*log2e) via v_exp_f32 (source modifiers fold the neg/abs)
    float e = __builtin_amdgcn_exp2f(-fabsf(d) * LOG2E);
    // log2(1 + e) via v_log_f32
    float l = __builtin_amdgcn_logf(1.0f + e);
    return fmaxf(d, 0.0f) + LN2 * l;
}

__global__ void __launch_bounds__(THREADS)
BradleyTerryModel_74337293959225_main(const float* __restrict__ W,
                                      const float* __restrict__ betas,
                                      float* __restrict__ partials) {
    __shared__ float sbrow[ROWS_PER_BLOCK];
    __shared__ float wsum[THREADS / 32];

    const int tid  = threadIdx.x;
    const int row0 = blockIdx.x * ROWS_PER_BLOCK;

    if (tid < ROWS_PER_BLOCK) sbrow[tid] = betas[row0 + tid];
    __syncthreads();

    // 4 independent accumulators -> independent v_fmac chains
    float ax = 0.0f, ay = 0.0f, az = 0.0f, aw = 0.0f;

    for (int cc = 0; cc < NDIM; cc += COLS_PER_CHUNK) {
        const int col = cc + tid * 4;
        const float4 bj = *reinterpret_cast<const float4*>(betas + col);

        // prefetch first row of the NEXT column chunk (global_prefetch_b8)
        if (cc + COLS_PER_CHUNK < NDIM) {
            __builtin_prefetch(W + (size_t)row0 * NDIM + col + COLS_PER_CHUNK, 0, 1);
        }

#pragma unroll
        for (int r = 0; r < ROWS_PER_BLOCK; ++r) {
            const float bi = sbrow[r];
            const float4 w = *reinterpret_cast<const float4*>(
                W + (size_t)(row0 + r) * NDIM + col);   // global_load_b128
            ax = fmaf(w.x, neg_logp(bi, bj.x), ax);
            ay = fmaf(w.y, neg_logp(bi, bj.y), ay);
            az = fmaf(w.z, neg_logp(bi, bj.z), az);
            aw = fmaf(w.w, neg_logp(bi, bj.w), aw);
        }
    }

    float acc = (ax + ay) + (az + aw);

    // ---- wave32 reduction with one V_WMMA_F32_16X16X4_F32 ----
    // A layout (16x4 f32): lanes 0-15 give A[m, K0/K1] = {acc, 0},
    //                      lanes 16-31 give A[m, K2/K3] = {acc, 0}.
    // B = all ones  =>  D[m, n] = acc[m] + acc[m+16]   (C = 0)
    v2f a; a.x = acc;  a.y = 0.0f;
    v2f b; b.x = 1.0f; b.y = 1.0f;
    v8f c = {};
    c = __builtin_amdgcn_wmma_f32_16x16x4_f32(
            /*neg_a=*/false, a, /*neg_b=*/false, b,
            /*c_mod=*/(short)0, c, /*reuse_a=*/false, /*reuse_b=*/false);
    // lanes 0-15 hold D rows M=0..7, lanes 16-31 hold rows M=8..15
    float h = ((c[0] + c[1]) + (c[2] + c[3])) + ((c[4] + c[5]) + (c[6] + c[7]));
    h += __shfl_xor(h, 16);   // combine the two half-row sums -> full wave total

    const int wid = tid >> 5;
    if ((tid & 31) == 0) wsum[wid] = h;
    __syncthreads();

    if (tid == 0) {
        float t = 0.0f;
#pragma unroll
        for (int w = 0; w < THREADS / 32; ++w) t += wsum[w];
        partials[blockIdx.x] = t;
    }
}

__global__ void __launch_bounds__(THREADS)
BradleyTerryModel_74337293959225_reduce(const float* __restrict__ W,
                                        const float* __restrict__ partials,
                                        float* __restrict__ out) {
    __shared__ float s[THREADS];
    const int tid = threadIdx.x;

    // sum block partials (fixed order -> deterministic)
    float v = 0.0f;
    for (int i = tid; i < GRID; i += THREADS) v += partials[i];

    // trace correction: full sum included diagonal terms W[i,i]*ln2
    float tr = 0.0f;
    for (int i = tid; i < NDIM; i += THREADS) tr += W[(size_t)i * (NDIM + 1)];

    s[tid] = v - LN2 * tr;
    __syncthreads();

#pragma unroll
    for (int step = THREADS / 2; step > 0; step >>= 1) {
        if (tid < step) s[tid] += s[tid + step];
        __syncthreads();
    }
    if (tid == 0) out[0] = s[0];   // = -loss
}

extern "C" void kernel_launch(void* const* d_in, const int* in_sizes, int n_in,
                              void* d_out, int out_size, void* d_ws, size_t ws_size,
                              hipStream_t stream) {
    const float* W     = (const float*)d_in[0];  // [8192*8192] fp32
    const float* betas = (const float*)d_in[1];  // [8192] fp32
    float* out      = (float*)d_out;
    float* partials = (float*)d_ws;              // GRID floats = 2 KB

    BradleyTerryModel_74337293959225_main<<<GRID, THREADS, 0, stream>>>(W, betas, partials);
    BradleyTerryModel_74337293959225_reduce<<<1, THREADS, 0, stream>>>(W, partials, out);
}